// GRU_12962211299468
// MI455X (gfx1250) — compile-verified
//
#include <hip/hip_runtime.h>
#include <math.h>

typedef __attribute__((ext_vector_type(2))) float v2f;
typedef __attribute__((ext_vector_type(8))) float v8f;

namespace {
constexpr int kB = 65536;
constexpr int kT = 9;
constexpr int kI = 57;
constexpr int kG = 6;        // 3*H gates
constexpr int kKsteps = 15;  // ceil(57/4), K padded to 60
constexpr int kWavesPerBlock = 2;
constexpr int kBlockThreads = kWavesPerBlock * 32;
constexpr int kBatchPerWave = 32;
constexpr int kBatchPerBlock = kWavesPerBlock * kBatchPerWave;  // 64
}  // namespace

__device__ __forceinline__ float fast_sigmoid(float v) {
  return 1.0f / (1.0f + __expf(-v));
}
__device__ __forceinline__ float fast_tanh(float v) {
  const float e = __expf(-2.0f * v);
  return (1.0f - e) / (1.0f + e);
}

__launch_bounds__(kBlockThreads)
__global__ void gru_fused_wmma(const float* __restrict__ x,
                               const float* __restrict__ W_ih,
                               const float* __restrict__ W_hh,
                               const float* __restrict__ b_ih,
                               const float* __restrict__ b_hh,
                               const float* __restrict__ fc_w,
                               const float* __restrict__ fc_b,
                               float* __restrict__ out) {
  // per-wave staging of gx: [t][row 0..31][gate 0..5], fp32
  __shared__ float gx_s[kWavesPerBlock * kT * kBatchPerWave * kG];

  const int tid = threadIdx.x;
  const int wave = tid >> 5;
  const int lane = tid & 31;
  const int nlo = lane & 15;   // N / M index within half-wave
  const int hi = lane >> 4;    // 0: K pair {0,1}; 1: K pair {2,3}
  const int b0 = blockIdx.x * kBatchPerBlock + wave * kBatchPerWave;
  float* wlds = gx_s + wave * (kT * kBatchPerWave * kG);

  // ---- B fragments: W_ih^T tiles (K padded 57->60, N padded 6->16) ----
  v2f bfrag[kKsteps];
#pragma unroll
  for (int ks = 0; ks < kKsteps; ++ks) {
    const int k = 4 * ks + 2 * hi;
    float bx = 0.0f, by = 0.0f;
    if (nlo < kG) {
      if (k < kI) bx = W_ih[nlo * kI + k];
      if (k + 1 < kI) by = W_ih[nlo * kI + k + 1];
    }
    v2f f;
    f.x = bx;
    f.y = by;
    bfrag[ks] = f;
  }

  const int rowStride = kT * kI;  // 513 floats per batch row
  const float* xr0 = x + (b0 + nlo) * rowStride;        // tile0: batches b0..b0+15
  const float* xr1 = x + (b0 + 16 + nlo) * rowStride;   // tile1: batches b0+16..b0+31

  // ---- input projection: gx[t] = x[t] @ W_ih^T via V_WMMA_F32_16X16X4_F32 ----
  for (int t = 0; t < kT; ++t) {
    const float* p0 = xr0 + t * kI;
    const float* p1 = xr1 + t * kI;
    if (t + 1 < kT) {
      __builtin_prefetch(p0 + kI, 0, 0);  // global_prefetch_b8: next timestep row
      __builtin_prefetch(p1 + kI, 0, 0);
    }
    v8f c0 = {};
    v8f c1 = {};
#pragma unroll
    for (int ks = 0; ks < kKsteps; ++ks) {
      const int k = 4 * ks + 2 * hi;
      v2f a0, a1;
      a0.x = (k < kI) ? p0[k] : 0.0f;
      a0.y = (k + 1 < kI) ? p0[k + 1] : 0.0f;
      a1.x = (k < kI) ? p1[k] : 0.0f;
      a1.y = (k + 1 < kI) ? p1[k + 1] : 0.0f;
      c0 = __builtin_amdgcn_wmma_f32_16x16x4_f32(false, a0, false, bfrag[ks],
                                                 (short)0, c0, false, false);
      c1 = __builtin_amdgcn_wmma_f32_16x16x4_f32(false, a1, false, bfrag[ks],
                                                 (short)0, c1, false, false);
    }
    // Scatter 6 live columns to LDS. D layout: VGPR j, lanes 0-15 -> M=j,
    // lanes 16-31 -> M=j+8; column N = nlo.
    if (nlo < kG) {
#pragma unroll
      for (int j = 0; j < 8; ++j) {
        const int m = j + 8 * hi;
        wlds[(t * kBatchPerWave + m) * kG + nlo] = c0[j];
        wlds[(t * kBatchPerWave + 16 + m) * kG + nlo] = c1[j];
      }
    }
  }

  __syncthreads();

  // ---- per-lane GRU recurrence (H=2) + FC, batch row = b0 + lane ----
  float wh[12], bi[6], bh[6];
#pragma unroll
  for (int i2 = 0; i2 < 12; ++i2) wh[i2] = W_hh[i2];
#pragma unroll
  for (int g = 0; g < 6; ++g) {
    bi[g] = b_ih[g];
    bh[g] = b_hh[g];
  }

  float h0 = 0.0f, h1 = 0.0f;
#pragma unroll
  for (int t = 0; t < kT; ++t) {
    const float* gt = wlds + (t * kBatchPerWave + lane) * kG;
    const float gh0 = fmaf(wh[0], h0, fmaf(wh[1], h1, bh[0]));
    const float gh1 = fmaf(wh[2], h0, fmaf(wh[3], h1, bh[1]));
    const float gh2 = fmaf(wh[4], h0, fmaf(wh[5], h1, bh[2]));
    const float gh3 = fmaf(wh[6], h0, fmaf(wh[7], h1, bh[3]));
    const float gh4 = fmaf(wh[8], h0, fmaf(wh[9], h1, bh[4]));
    const float gh5 = fmaf(wh[10], h0, fmaf(wh[11], h1, bh[5]));
    const float r0 = fast_sigmoid(gt[0] + bi[0] + gh0);
    const float r1 = fast_sigmoid(gt[1] + bi[1] + gh1);
    const float z0 = fast_sigmoid(gt[2] + bi[2] + gh2);
    const float z1 = fast_sigmoid(gt[3] + bi[3] + gh3);
    const float n0 = fast_tanh(fmaf(r0, gh4, gt[4] + bi[4]));
    const float n1 = fast_tanh(fmaf(r1, gh5, gt[5] + bi[5]));
    h0 = fmaf(z0, h0 - n0, n0);  // (1-z)*n + z*h
    h1 = fmaf(z1, h1 - n1, n1);
  }
  out[b0 + lane] = fmaf(fc_w[0], h0, fmaf(fc_w[1], h1, fc_b[0]));
}

extern "C" void kernel_launch(void* const* d_in, const int* in_sizes, int n_in,
                              void* d_out, int out_size, void* d_ws, size_t ws_size,
                              hipStream_t stream) {
  const float* x    = (const float*)d_in[0];
  const float* W_ih = (const float*)d_in[1];
  const float* W_hh = (const float*)d_in[2];
  const float* b_ih = (const float*)d_in[3];
  const float* b_hh = (const float*)d_in[4];
  const float* fc_w = (const float*)d_in[5];
  const float* fc_b = (const float*)d_in[6];
  float* out = (float*)d_out;

  dim3 grid(kB / kBatchPerBlock);   // 1024 blocks
  dim3 block(kBlockThreads);        // 64 threads = 2 waves
  hipLaunchKernelGGL(gru_fused_wmma, grid, block, 0, stream,
                     x, W_ih, W_hh, b_ih, b_hh, fc_w, fc_b, out);
}